// ControlledODE_60258391162956
// MI455X (gfx1250) — compile-verified
//
#include <hip/hip_runtime.h>

// v: (B=32, C=2, 128, 128, Z=32) f32 ; A: (2, 64, 64, 32, 32) f32
// out = zeros_like(v); out[:, :, 32:96, 32:96, :] = complex matvec per mode.
#define BATCH 32
#define ZDIM  32
#define DXL   128
#define DYL   128
#define NMODE 64
#define CROP  32
#define MODE_BLOCKS (NMODE * NMODE)   // 4096
#define FILL_BLOCKS 2048

typedef __attribute__((ext_vector_type(2))) float v2f;
typedef __attribute__((ext_vector_type(4))) float v4f;
typedef __attribute__((ext_vector_type(8))) float v8f;
typedef __attribute__((vector_size(16)))    int   v4i;   // matches builtin param

#if __has_builtin(__builtin_amdgcn_global_load_async_to_lds_b128) && \
    __has_builtin(__builtin_amdgcn_global_load_async_to_lds_b32)  && \
    __has_builtin(__builtin_amdgcn_s_wait_asynccnt)
#define HAVE_ASYNC_LDS 1
#endif

// ---------------------------------------------------------------------------
// Fused kernel. Blocks [0, 4096): one (mx,my) mode each -> 4 real 32x32x32
// GEMMs on v_wmma_f32_16x16x4_f32. Blocks [4096, 6144): zero-fill every
// float4 of the output EXCEPT the 64x64 center window (compute writes those).
// Staging uses GLOBAL_LOAD_ASYNC_TO_LDS when available: A via per-lane b128
// (512B per wave instruction), vT via per-lane b32 scatter that performs the
// transpose while keeping 128B-coalesced global reads. Each wave does
// s_wait_asynccnt 0 before the workgroup barrier.
// ---------------------------------------------------------------------------
__global__ __launch_bounds__(256)
void fno_fused(const float* __restrict__ v,
               const float* __restrict__ A,
               float* __restrict__ out, int n4)
{
    const int t = threadIdx.x;

    if (blockIdx.x >= MODE_BLOCKS) {
        // ---------------- zero-fill role (hole-punched) ----------------
        v4f* o4 = (v4f*)out;
        const v4f z = {0.f, 0.f, 0.f, 0.f};
        int idx = (blockIdx.x - MODE_BLOCKS) * 256 + t;
        for (int i4 = idx; i4 < n4; i4 += FILL_BLOCKS * 256) {
            int y = (i4 >> 3) & 127;          // 8 float4 per z-row
            int x = (i4 >> 10) & 127;         // 1024 float4 per x slice
            bool center = ((unsigned)(x - CROP) < (unsigned)NMODE) &
                          ((unsigned)(y - CROP) < (unsigned)NMODE);
            if (!center) __builtin_nontemporal_store(z, &o4[i4]);
        }
        return;
    }

    // ---------------- mode-GEMM role ----------------
    const int mx = blockIdx.x >> 6;
    const int my = blockIdx.x & 63;
    const int X  = CROP + mx;
    const int Y  = CROP + my;

    __shared__ float sA0[ZDIM][36];            // A[0][mx][my][i][j]
    __shared__ float sA1[ZDIM][36];            // A[1][mx][my][i][j]
    __shared__ float sV0[ZDIM][BATCH + 1];     // vT[j][b], c=0
    __shared__ float sV1[ZDIM][BATCH + 1];     // vT[j][b], c=1
    __shared__ float sOut[2][BATCH][36];       // [part][b][i]

    const float* Abase0 = A + (size_t)((0 * NMODE + mx) * NMODE + my) * (ZDIM * ZDIM);
    const float* Abase1 = A + (size_t)((1 * NMODE + mx) * NMODE + my) * (ZDIM * ZDIM);

#ifdef HAVE_ASYNC_LDS
    // ---- async stage A: one b128 per lane -> 512B per wave instruction ----
    {
        int i  = t >> 3;                  // 0..31 (row)
        int j4 = (t & 7) << 2;            // float4 slot
        __builtin_amdgcn_global_load_async_to_lds_b128(
            (v4i*)(Abase0 + i * ZDIM + j4), (v4i*)&sA0[i][j4], 0, 0);
        __builtin_amdgcn_global_load_async_to_lds_b128(
            (v4i*)(Abase1 + i * ZDIM + j4), (v4i*)&sA1[i][j4], 0, 0);
    }
    // ---- async stage vT: per-lane b32 scatter does the transpose;
    //      global side stays 128B-contiguous (j = lane) ----
    {
        int j = t & 31;
#pragma unroll
        for (int p = 0; p < 4; ++p) {
            int b = (t >> 5) + (p << 3);  // wave-uniform batch row
            const float* v0g = v + ((((size_t)(b * 2 + 0) * DXL + X) * DYL + Y) << 5) + j;
            const float* v1g = v + ((((size_t)(b * 2 + 1) * DXL + X) * DYL + Y) << 5) + j;
            __builtin_amdgcn_global_load_async_to_lds_b32(
                (int*)v0g, (int*)&sV0[j][b], 0, 0);
            __builtin_amdgcn_global_load_async_to_lds_b32(
                (int*)v1g, (int*)&sV1[j][b], 0, 0);
        }
    }
    __builtin_amdgcn_s_wait_asynccnt(0);  // this wave's LDS data resident
    __syncthreads();                      // all waves' data resident
#else
    // ---- fallback: synchronous wide staging ----
    {
        const v4f* A0g = (const v4f*)Abase0;
        const v4f* A1g = (const v4f*)Abase1;
        int i  = t >> 3;
        int q  = t & 7;
        int j4 = q << 2;
        v4f a0 = __builtin_nontemporal_load(&A0g[(i << 3) + q]);
        v4f a1 = __builtin_nontemporal_load(&A1g[(i << 3) + q]);
        *(v4f*)&sA0[i][j4] = a0;
        *(v4f*)&sA1[i][j4] = a1;
    }
    {
        int b  = t >> 3;
        int j4 = (t & 7) << 2;
        const float* v0g = v + ((((size_t)(b * 2 + 0) * DXL + X) * DYL + Y) << 5);
        const float* v1g = v + ((((size_t)(b * 2 + 1) * DXL + X) * DYL + Y) << 5);
        v4f q0 = __builtin_nontemporal_load((const v4f*)(v0g + j4));
        v4f q1 = __builtin_nontemporal_load((const v4f*)(v1g + j4));
        sV0[j4 + 0][b] = q0.x; sV0[j4 + 1][b] = q0.y;
        sV0[j4 + 2][b] = q0.z; sV0[j4 + 3][b] = q0.w;
        sV1[j4 + 0][b] = q1.x; sV1[j4 + 1][b] = q1.y;
        sV1[j4 + 2][b] = q1.z; sV1[j4 + 3][b] = q1.w;
    }
    __syncthreads();
#endif

    // ---- per-wave 16x16 tile: (part, i-tile, b-tile) ----
    const int lane = t & 31;
    const int w    = t >> 5;
    const int part = w >> 2;            // 0 = real, 1 = imag
    const int mi   = (w >> 1) & 1;
    const int nb   = w & 1;
    const int half = lane >> 4;
    const int irow = mi * 16 + (lane & 15);
    const int bcol = nb * 16 + (lane & 15);

    const float (*Af)[36] = part ? sA1 : sA0;   // term with v0
    const float (*As)[36] = part ? sA0 : sA1;   // term with v1

    v8f acc0 = {0.f, 0.f, 0.f, 0.f, 0.f, 0.f, 0.f, 0.f};
    v8f acc1 = {0.f, 0.f, 0.f, 0.f, 0.f, 0.f, 0.f, 0.f};

#pragma unroll
    for (int k = 0; k < 8; ++k) {
        // f32 16x16x4: VGPR0 = K base+0 (lanes 0-15)/base+2 (16-31), VGPR1 = +1/+3
        int jb = 4 * k + 2 * half;

        v2f fa0 = { Af[irow][jb], Af[irow][jb + 1] };
        v2f fb0 = { sV0[jb][bcol], sV0[jb + 1][bcol] };
        acc0 = __builtin_amdgcn_wmma_f32_16x16x4_f32(
            false, fa0, false, fb0, (short)0, acc0, false, false);

        v2f fa1 = { As[irow][jb], As[irow][jb + 1] };
        v2f fb1 = { sV1[jb][bcol], sV1[jb + 1][bcol] };
        acc1 = __builtin_amdgcn_wmma_f32_16x16x4_f32(
            false, fa1, false, fb1, (short)0, acc1, false, false);
    }

    // real: acc0 - acc1 ; imag: acc0 + acc1
    const float sgn = part ? 1.0f : -1.0f;
#pragma unroll
    for (int r = 0; r < 8; ++r) {
        int i = mi * 16 + half * 8 + r;     // D layout: VGPR r -> M=r / M=r+8
        sOut[part][bcol][i] = acc0[r] + sgn * acc1[r];
    }
    __syncthreads();

    // ---- writeback: b128 LDS reads + 16B NT stores, 128B-coalesced rows ----
#pragma unroll
    for (int p = 0; p < 2; ++p) {
        int idx4 = t + 256 * p;             // 512 float4s total
        int i4   = (idx4 & 7) << 2;
        int b2   = (idx4 >> 3) & 31;
        int pt   = idx4 >> 8;
        v4f d = *(const v4f*)&sOut[pt][b2][i4];
        size_t off = ((((size_t)(b2 * 2 + pt) * DXL + X) * DYL + Y) << 5) + i4;
        __builtin_nontemporal_store(d, (v4f*)(out + off));
    }
}

// ---------------------------------------------------------------------------
extern "C" void kernel_launch(void* const* d_in, const int* in_sizes, int n_in,
                              void* d_out, int out_size, void* d_ws, size_t ws_size,
                              hipStream_t stream)
{
    // inputs: d_in[0] = t (scalar, unused), d_in[1] = v, d_in[2] = A
    const float* v = (const float*)d_in[1];
    const float* A = (const float*)d_in[2];
    float* out     = (float*)d_out;

    int n4 = out_size >> 2;
    fno_fused<<<MODE_BLOCKS + FILL_BLOCKS, 256, 0, stream>>>(v, A, out, n4);
}